// Net_12833362280838
// MI455X (gfx1250) — compile-verified
//
#include <hip/hip_runtime.h>

typedef __attribute__((ext_vector_type(2))) float v2f;
typedef __attribute__((ext_vector_type(8))) float v8f;
typedef __attribute__((ext_vector_type(4))) unsigned int u32x4;
typedef __attribute__((ext_vector_type(8))) int i32x8;
typedef __attribute__((ext_vector_type(4))) int i32x4;

#define IN_C    128
#define HID_C   128
#define OUT_C   64

__device__ __forceinline__ void atomAddF(float* p, float v) {
    // native global_atomic_add_f32 (no CAS loop)
    unsafeAtomicAdd(p, v);
}

// ---------------------------------------------------------------------------
// Tensor Data Mover: 2-D tile load global -> LDS (fp32 elements).
// Packs D# group0 (128b) and group1 (256b) per CDNA5 ISA 8.3/8.4; groups 2/3
// unused (2-D tensor). dim0/tile0 = inner (contiguous) extent in elements,
// dim1/tile1 = rows, stride0 = row stride in elements.
// ---------------------------------------------------------------------------
__device__ __forceinline__ void tdm_load_2d(unsigned lds_off, const void* gptr,
                                            unsigned dim0, unsigned dim1,
                                            unsigned tile0, unsigned tile1,
                                            unsigned long long stride0) {
    unsigned long long ga = (unsigned long long)(uintptr_t)gptr;
    u32x4 g0;
    g0[0] = 1u;                                   // count=1, user descriptor
    g0[1] = lds_off;                              // lds_addr (bytes)
    g0[2] = (unsigned)(ga & 0xFFFFFFFFu);         // global_addr[31:0]
    g0[3] = (unsigned)((ga >> 32) & 0x01FFFFFFu)  // global_addr[56:32]
          | (2u << 30);                           // type = 2 ("image")
    i32x8 g1;
    g1[0] = (int)(2u << 16);                      // data_size = 4 bytes
    g1[1] = (int)((dim0 & 0xFFFFu) << 16);                          // dim0[15:0]
    g1[2] = (int)(((dim0 >> 16) & 0xFFFFu) | ((dim1 & 0xFFFFu) << 16));
    g1[3] = (int)(((dim1 >> 16) & 0xFFFFu) | ((tile0 & 0xFFFFu) << 16));
    g1[4] = (int)(tile1 & 0xFFFFu);               // tile_dim1; tile_dim2 = 0
    g1[5] = (int)(stride0 & 0xFFFFFFFFull);       // tensor_dim0_stride[31:0]
    g1[6] = (int)((stride0 >> 32) & 0xFFFFull);   // stride0[47:32]
    g1[7] = 0;
    i32x4 z4 = {0, 0, 0, 0};                      // groups 2/3 disabled (2-D)
    i32x8 z8 = {0, 0, 0, 0, 0, 0, 0, 0};
    __builtin_amdgcn_tensor_load_to_lds(g0, g1, z4, z4, z8, 0);
}

// ---------------------------------------------------------------------------
// GEMM: H[N,M] = act(A[N,K]) @ W[K,M] with V_WMMA_F32_16X16X4_F32.
// One block = 16-row strip; wave w handles col tile w. W (K*M) and the A strip
// (16*K) are staged into LDS by the Tensor Data Mover; WMMA operands are read
// from LDS only. blockDim.x = 32*(M/16); dynamic LDS = (K*M + 16*K)*4 bytes.
// ---------------------------------------------------------------------------
template <int K, int M, bool RELU_IN>
__global__ void gemm_wmma_f32(const float* __restrict__ A,
                              const float* __restrict__ W,
                              float* __restrict__ H, int N) {
    extern __shared__ float smem[];
    float* sW = smem;            // K*M floats, row-major (same as global W)
    float* sA = smem + K * M;    // 16*K floats, row-major strip

    const int lane = threadIdx.x & 31;
    const int wave = threadIdx.x >> 5;
    const int row0 = blockIdx.x * 16;

    if (wave == 0) {
        const unsigned ldsW = (unsigned)(uintptr_t)sW;
        const unsigned ldsA = (unsigned)(uintptr_t)sA;
        // whole weight matrix: K rows of M floats, contiguous
        tdm_load_2d(ldsW, W, (unsigned)M, (unsigned)K, (unsigned)M, (unsigned)K,
                    (unsigned long long)M);
        // 16-row strip of A; tensor_dim1 = remaining rows so OOB rows zero-fill
        unsigned rem = (unsigned)(N - row0);
        tdm_load_2d(ldsA, A + (size_t)row0 * K, (unsigned)K, rem, (unsigned)K, 16u,
                    (unsigned long long)K);
        __builtin_amdgcn_s_wait_tensorcnt(0);
    }
    __syncthreads();

    const int m  = lane & 15;    // row (A) / col (B) within 16x16 tile
    const int hi = lane >> 4;    // selects K pair {0,1} vs {2,3}
    const float* apl = sA + m * K;
    const int bcol = wave * 16 + m;

    v8f c = {};
#pragma unroll 4
    for (int kk = 0; kk < K; kk += 4) {
        const int ka = kk + 2 * hi;
        // A-matrix fragment: VGPR0 = K=ka, VGPR1 = K=ka+1 for row m (ds b64)
        v2f a = *(const v2f*)(apl + ka);
        if (RELU_IN) {
            a.x = fmaxf(a.x, 0.0f);
            a.y = fmaxf(a.y, 0.0f);
        }
        // B-matrix fragment: rows K=ka, K=ka+1 at column bcol
        v2f b;
        b.x = sW[ka * M + bcol];
        b.y = sW[(ka + 1) * M + bcol];
        c = __builtin_amdgcn_wmma_f32_16x16x4_f32(
                /*neg_a=*/false, a, /*neg_b=*/false, b,
                /*c_mod=*/(short)0, c, /*reuse_a=*/false, /*reuse_b=*/false);
    }

    // C/D layout: VGPR r -> row (row0 + r + 8*hi), col (wave*16 + m)
    float* hp = H + ((size_t)row0 + 8 * hi) * M + wave * 16 + m;
    if (row0 + 16 <= N) {
        // full tile: straight-line clause of 8 stores, immediate offsets
#pragma unroll
        for (int r = 0; r < 8; ++r) hp[(size_t)r * M] = c[r];
    } else {
        // ragged tail tile (not taken when N % 16 == 0)
#pragma unroll
        for (int r = 0; r < 8; ++r) {
            if (row0 + 8 * hi + r < N) hp[(size_t)r * M] = c[r];
        }
    }
}

// ---------------------------------------------------------------------------
// Degree / normalization
// ---------------------------------------------------------------------------
__global__ void deg_init(float* __restrict__ deg, int N) {
    int t = blockIdx.x * blockDim.x + threadIdx.x;
    if (t < N) deg[t] = 1.0f;             // self-loop
}

__global__ void deg_count(const int* __restrict__ dst, float* __restrict__ deg, int E) {
    int t = blockIdx.x * blockDim.x + threadIdx.x;
    if (t < E) atomAddF(&deg[dst[t]], 1.0f);
}

__global__ void deg_rsqrt(float* __restrict__ deg, int N) {
    int t = blockIdx.x * blockDim.x + threadIdx.x;
    if (t < N) deg[t] = rsqrtf(deg[t]);
}

// ---------------------------------------------------------------------------
// z[i] = bias + h[i] * dinv[i]^2  (self-loop message + bias); float4 / thread
// ---------------------------------------------------------------------------
__global__ void self_init(const float* __restrict__ h, const float* __restrict__ dinv,
                          const float* __restrict__ bias, float* __restrict__ z,
                          int total4, int cShift) {
    int t = blockIdx.x * blockDim.x + threadIdx.x;
    if (t >= total4) return;
    const int node = t >> cShift;
    const int g    = t & ((1 << cShift) - 1);
    const float d  = dinv[node];
    const float s  = d * d;
    const float4 hv = ((const float4*)h)[t];
    const float4 bv = ((const float4*)bias)[g];
    float4 o;
    o.x = bv.x + hv.x * s;
    o.y = bv.y + hv.y * s;
    o.z = bv.z + hv.z * s;
    o.w = bv.w + hv.w * s;
    ((float4*)z)[t] = o;
}

// ---------------------------------------------------------------------------
// Edge scatter: z[dst] += h[src] * dinv[src]*dinv[dst]; thread = 4 channels
// ---------------------------------------------------------------------------
__global__ void edge_scatter(const float* __restrict__ h, const float* __restrict__ dinv,
                             const int* __restrict__ src, const int* __restrict__ dst,
                             float* __restrict__ z, int total, int perShift, int C) {
    int t = blockIdx.x * blockDim.x + threadIdx.x;
    if (t >= total) return;
    const int e  = t >> perShift;
    const int c4 = (t & ((1 << perShift) - 1)) << 2;
    const int s = src[e];
    const int d = dst[e];
    const float norm = dinv[s] * dinv[d];
    const float4 hv = *(const float4*)(h + (size_t)s * C + c4);
    float* zp = z + (size_t)d * C + c4;
    atomAddF(zp + 0, hv.x * norm);
    atomAddF(zp + 1, hv.y * norm);
    atomAddF(zp + 2, hv.z * norm);
    atomAddF(zp + 3, hv.w * norm);
}

// ---------------------------------------------------------------------------
// Decode: out[e] = dot(z[a[e]], z[b[e]]) over OUT_C channels
// ---------------------------------------------------------------------------
__global__ void decode_dot(const float* __restrict__ z,
                           const int* __restrict__ ia, const int* __restrict__ ib,
                           float* __restrict__ out, int L) {
    int t = blockIdx.x * blockDim.x + threadIdx.x;
    if (t >= L) return;
    const float4* za = (const float4*)(z + (size_t)ia[t] * OUT_C);
    const float4* zb = (const float4*)(z + (size_t)ib[t] * OUT_C);
    float acc = 0.0f;
#pragma unroll
    for (int i = 0; i < OUT_C / 4; ++i) {
        const float4 a = za[i];
        const float4 b = zb[i];
        acc += a.x * b.x + a.y * b.y + a.z * b.z + a.w * b.w;
    }
    out[t] = acc;
}

// ---------------------------------------------------------------------------
extern "C" void kernel_launch(void* const* d_in, const int* in_sizes, int n_in,
                              void* d_out, int out_size, void* d_ws, size_t ws_size,
                              hipStream_t stream) {
    const float* x  = (const float*)d_in[0];
    const float* W1 = (const float*)d_in[1];
    const float* b1 = (const float*)d_in[2];
    const float* W2 = (const float*)d_in[3];
    const float* b2 = (const float*)d_in[4];
    const float* W3 = (const float*)d_in[5];
    const float* b3 = (const float*)d_in[6];
    const int*   edge_index       = (const int*)d_in[7];
    const int*   edge_label_index = (const int*)d_in[8];
    float*       out = (float*)d_out;

    const int N = in_sizes[0] / IN_C;
    const int E = in_sizes[7] / 2;
    const int L = in_sizes[8] / 2;

    const int* e_src = edge_index;
    const int* e_dst = edge_index + E;
    const int* l_a   = edge_label_index;
    const int* l_b   = edge_label_index + L;

    // workspace: dinv | bufA (h, N*128) | bufB (z, N*128)
    float* dinv = (float*)d_ws;
    float* bufA = dinv + (((size_t)N + 1023) & ~(size_t)1023);
    float* bufB = bufA + (size_t)N * HID_C;

    const int TB = 256;
    const int rowTiles = (N + 15) / 16;
    const size_t sh128 = (size_t)(HID_C * HID_C + 16 * HID_C) * sizeof(float); // 72 KB
    const size_t sh64  = (size_t)(HID_C * OUT_C + 16 * HID_C) * sizeof(float); // 40 KB

    // --- degrees & rsqrt normalization ---
    deg_init <<<(N + TB - 1) / TB, TB, 0, stream>>>(dinv, N);
    deg_count<<<(E + TB - 1) / TB, TB, 0, stream>>>(e_dst, dinv, E);
    deg_rsqrt<<<(N + TB - 1) / TB, TB, 0, stream>>>(dinv, N);

    // --- layer 1: h1 = x @ W1 ; z1 = scatter(h1) + b1 ---
    gemm_wmma_f32<IN_C, HID_C, false><<<rowTiles, 32 * (HID_C / 16), sh128, stream>>>(x, W1, bufA, N);
    {
        int total4 = N * (HID_C / 4);
        self_init<<<(total4 + TB - 1) / TB, TB, 0, stream>>>(bufA, dinv, b1, bufB, total4, 5);
        int total = E * (HID_C / 4);
        edge_scatter<<<(total + TB - 1) / TB, TB, 0, stream>>>(bufA, dinv, e_src, e_dst, bufB, total, 5, HID_C);
    }

    // --- layer 2: h2 = relu(z1) @ W2 ; z2 = scatter(h2) + b2 ---
    gemm_wmma_f32<HID_C, HID_C, true><<<rowTiles, 32 * (HID_C / 16), sh128, stream>>>(bufB, W2, bufA, N);
    {
        int total4 = N * (HID_C / 4);
        self_init<<<(total4 + TB - 1) / TB, TB, 0, stream>>>(bufA, dinv, b2, bufB, total4, 5);
        int total = E * (HID_C / 4);
        edge_scatter<<<(total + TB - 1) / TB, TB, 0, stream>>>(bufA, dinv, e_src, e_dst, bufB, total, 5, HID_C);
    }

    // --- layer 3: h3 = relu(z2) @ W3 ; z3 = scatter(h3) + b3 (no relu) ---
    gemm_wmma_f32<HID_C, OUT_C, true><<<rowTiles, 32 * (OUT_C / 16), sh64, stream>>>(bufB, W3, bufA, N);
    {
        int total4 = N * (OUT_C / 4);
        self_init<<<(total4 + TB - 1) / TB, TB, 0, stream>>>(bufA, dinv, b3, bufB, total4, 4);
        int total = E * (OUT_C / 4);
        edge_scatter<<<(total + TB - 1) / TB, TB, 0, stream>>>(bufA, dinv, e_src, e_dst, bufB, total, 4, OUT_C);
    }

    // --- decode ---
    decode_dot<<<(L + TB - 1) / TB, TB, 0, stream>>>(bufB, l_a, l_b, out, L);
}